// PostProcess_90752658964736
// MI455X (gfx1250) — compile-verified
//
#include <hip/hip_runtime.h>
#include <hip/hip_bf16.h>
#include <stdint.h>

// Problem constants (from reference)
#define Bx   32
#define Kk   17
#define Hh   192
#define Ww   192
#define Cc   19
#define Pp   20
#define Ss   10
#define NPIX (Hh*Ww)          // 36864 pixels per map (144 KB fp32)
#define T1   512              // kernel-1 block size (16 wave32)
#define CHUNK (NPIX/T1)       // 72 strided elements per thread
#define T2   416              // kernel-2 block size (13 wave32)
#define PEAK_TH 0.1f
#define PAF_TH  0.05f
#define CNT_TH  0.8f

// TDM covers rows 0..95 (one 6-row slab per wave); per-lane async covers 96..191
#define TDM_ELEMS   1152u     // 6 rows * 192 cols, fp32
#define TDM_BYTES   4608u
#define HALF_FLOATS 18432     // 96 rows * 192
#define HALF_VEC4   4608      // HALF_FLOATS / 4

__constant__ int SKA[Cc] = {15,13,16,14,11, 5, 6, 5, 5, 6, 7, 8, 1, 0, 0, 1, 2, 3, 4};
__constant__ int SKB[Cc] = {13,11,14,12,12,11,12, 6, 7, 8, 9,10, 2, 1, 2, 3, 4, 5, 6};

// LDS byte offset of a generic pointer to shared memory: flat LDS aperture
// keeps the workgroup-relative byte offset in addr[31:0] (ISA 10.2).
__device__ __forceinline__ uint32_t lds_off(const void* p) {
  return (uint32_t)(uintptr_t)p;
}

// ---------------------------------------------------------------------------
// Kernel 1: per-(b,k) peak detection + top-20 with subpixel refinement.
// One block owns one 192x192 map (294912 B LDS <= 320 KB/WGP).
// Phase A uses BOTH CDNA5 copy engines: TDM tensor_load_to_lds (TENSORcnt)
// for the top half and global_load_async_to_lds_b128 (ASYNCcnt) for the rest.
// ---------------------------------------------------------------------------
__global__ __launch_bounds__(T1) void peaks_kernel(const float* __restrict__ heat,
                                                   float* __restrict__ out_peaks, // B*K*P*3
                                                   float* __restrict__ raw)       // B*K*P*4
{
  extern __shared__ float smem[];
  float* heat_s  = smem;          // NPIX floats
  float* score_s = smem + NPIX;   // NPIX floats
  __shared__ float red_s[T1/32];
  __shared__ int   red_i[T1/32];
  __shared__ int   winner;

  const int tid = threadIdx.x;
  const int map = blockIdx.x;                 // b*K + k
  const float* g = heat + (size_t)map * NPIX;

  // ---- Phase A1: TDM DMA, one 6-row slab per wave (rows 0..95).
  {
    const uint32_t wave = (uint32_t)__builtin_amdgcn_readfirstlane(threadIdx.x >> 5);
    const uint64_t gaddr = (uint64_t)(uintptr_t)g + (uint64_t)wave * TDM_BYTES;
    const uint32_t glo = (uint32_t)__builtin_amdgcn_readfirstlane((int)(uint32_t)gaddr);
    uint32_t ghi = (uint32_t)__builtin_amdgcn_readfirstlane((int)(uint32_t)(gaddr >> 32));
    ghi = (ghi & 0x01FFFFFFu) | 0x80000000u;   // global_addr[56:32] | D#.type=2
    const uint32_t lds = (uint32_t)__builtin_amdgcn_readfirstlane(
        (int)(lds_off(heat_s) + wave * TDM_BYTES));
    // D# group0 (s[40:43]): count=1 | lds_addr | global_addr | type=2
    // D# group1 (s[48:55]): data_size=4B, tensor_dim0=tile_dim0=1152,
    //                       tensor_dim1=tile_dim1=1, dim0_stride=1152,
    //                       no multicast/pad/iterate/gather/atomic-barrier.
    asm volatile(
      "s_mov_b32 s40, 1\n\t"
      "s_mov_b32 s41, %0\n\t"
      "s_mov_b32 s42, %1\n\t"
      "s_mov_b32 s43, %2\n\t"
      "s_mov_b32 s48, 0x20000\n\t"
      "s_mov_b32 s49, 0x04800000\n\t"
      "s_mov_b32 s50, 0x10000\n\t"
      "s_mov_b32 s51, 0x04800000\n\t"
      "s_mov_b32 s52, 1\n\t"
      "s_mov_b32 s53, 0x480\n\t"
      "s_mov_b32 s54, 0\n\t"
      "s_mov_b32 s55, 0\n\t"
      "tensor_load_to_lds s[40:43], s[48:55]\n\t"
      :: "s"(lds), "s"(glo), "s"(ghi)
      : "s40","s41","s42","s43","s48","s49","s50","s51","s52","s53","s54","s55",
        "memory");
  }

  // ---- Phase A2: per-lane async global->LDS b128 for rows 96..191.
  {
    const uint32_t lbase = lds_off(heat_s);
    #pragma unroll
    for (int j = 0; j < HALF_VEC4 / T1; ++j) {  // 9 x 16B per thread
      const int v = HALF_VEC4 + tid + j * T1;   // float4 index into the map
      const uint64_t ga = (uint64_t)(uintptr_t)(g + 4*v);
      const uint32_t la = lbase + (uint32_t)v * 16u;
      asm volatile("global_load_async_to_lds_b128 %0, %1, off"
                   :: "v"(la), "v"(ga) : "memory");
    }
    asm volatile("s_wait_asynccnt 0" ::: "memory");
  }
  __builtin_amdgcn_s_wait_tensorcnt(0);
  __syncthreads();

  // ---- Phase B: 3x3 local-max score map (SAME padding => clipped window)
  for (int j = 0; j < CHUNK; ++j) {
    const int idx = tid + j * T1;
    const int y = idx / Ww, x = idx - y * Ww;
    const float h = heat_s[idx];
    float m = h;
    const int x0 = (x > 0) ? -1 : 0, x1 = (x < Ww-1) ? 1 : 0;
    const int y0 = (y > 0) ? -1 : 0, y1 = (y < Hh-1) ? 1 : 0;
    for (int dy = y0; dy <= y1; ++dy)
      for (int dx = x0; dx <= x1; ++dx)
        m = fmaxf(m, heat_s[idx + dy*Ww + dx]);
    score_s[idx] = (h == m && h > PEAK_TH) ? h : -1.0f;
  }
  __syncthreads();

  // ---- Phase C: 20 rounds of block-argmax (tie -> lowest index, matches top_k)
  float bs = -1e30f; int bi = 0x7fffffff;
  for (int j = 0; j < CHUNK; ++j) {
    const int idx = tid + j * T1;
    const float s = score_s[idx];
    if (s > bs || (s == bs && idx < bi)) { bs = s; bi = idx; }
  }

  const int wid  = tid >> 5;
  const int lane = tid & 31;
  const int obase = map * Pp;

  for (int it = 0; it < Pp; ++it) {
    // wave32 shuffle reduction of cached (score, idx)
    float s = bs; int id = bi;
    #pragma unroll
    for (int off = 16; off; off >>= 1) {
      const float s2 = __shfl_down(s,  off, 32);
      const int   i2 = __shfl_down(id, off, 32);
      if (s2 > s || (s2 == s && i2 < id)) { s = s2; id = i2; }
    }
    if (lane == 0) { red_s[wid] = s; red_i[wid] = id; }
    __syncthreads();

    if (tid == 0) {
      float ws = red_s[0]; int widx = red_i[0];
      for (int w = 1; w < T1/32; ++w) {
        const float s2 = red_s[w]; const int i2 = red_i[w];
        if (s2 > ws || (s2 == ws && i2 < widx)) { ws = s2; widx = i2; }
      }
      winner = widx;
      score_s[widx] = -3.0f;                 // remove from further rounds

      const int y = widx / Ww, x = widx - y * Ww;
      float ddx = 0.0f, ddy = 0.0f;
      if (x > 0 && x < Ww-1 && y > 0 && y < Hh-1) {
        const float h = heat_s[widx];
        const float r = heat_s[widx+1],  l = heat_s[widx-1];
        const float d = heat_s[widx+Ww], u = heat_s[widx-Ww];
        ddx = 0.5f * (r - l); const float dxx = r + l - 2.0f*h;
        if (dxx != 0.0f) ddx = ddx / -dxx;
        ddy = 0.5f * (d - u); const float dyy = d + u - 2.0f*h;
        if (dyy != 0.0f) ddy = ddy / -dyy;
      }
      const float pxf = (float)x + ddx, pyf = (float)y + ddy;
      const float vm  = (ws > PEAK_TH) ? 1.0f : 0.0f;
      float* po = out_peaks + (size_t)(obase + it) * 3;
      po[0] = pxf * vm; po[1] = pyf * vm; po[2] = ws * vm;
      float* ro = raw + (size_t)(obase + it) * 4;   // unmasked, for kernel 2
      ro[0] = pxf; ro[1] = pyf; ro[2] = ws; ro[3] = vm;
    }
    __syncthreads();

    // only the thread whose chunk held the winner rescans
    if ((winner & (T1-1)) == tid) {
      bs = -1e30f; bi = 0x7fffffff;
      for (int j = 0; j < CHUNK; ++j) {
        const int idx = tid + j * T1;
        const float sv = score_s[idx];
        if (sv > bs || (sv == bs && idx < bi)) { bs = sv; bi = idx; }
      }
    }
  }
}

// ---------------------------------------------------------------------------
// Kernel 2: per-(b,c) edge scoring over 20x20 peak pairs, 10 PAF samples each.
// ---------------------------------------------------------------------------
__global__ __launch_bounds__(T2) void conn_kernel(const float* __restrict__ paf,
                                                  const float* __restrict__ raw,
                                                  float* __restrict__ out_conn)
{
  __shared__ float pa[Pp*4];
  __shared__ float pb[Pp*4];

  const int tid = threadIdx.x;
  const int bc  = blockIdx.x;               // b*C + c
  const int b   = bc / Cc, c = bc - b * Cc;
  const float* ra = raw + (size_t)((b*Kk + SKA[c]) * Pp) * 4;
  const float* rb = raw + (size_t)((b*Kk + SKB[c]) * Pp) * 4;
  if (tid < Pp*4)            pa[tid]        = ra[tid];
  else if (tid < Pp*8)       pb[tid - Pp*4] = rb[tid - Pp*4];

  const float* pafx = paf + (size_t)(b*2*Cc + 2*c) * NPIX;
  const float* pafy = pafx + NPIX;

  // warm both PAF channels into cache (global_prefetch_b8), one line per iter
  for (int off = tid * 32; off < NPIX; off += T2 * 32) {
    __builtin_prefetch(pafx + off, 0, 0);
    __builtin_prefetch(pafy + off, 0, 0);
  }
  __syncthreads();

  if (tid < Pp*Pp) {
    const int i = tid / Pp, j = tid - i * Pp;
    const float ax = pa[i*4+0], ay = pa[i*4+1], sa = pa[i*4+2], va = pa[i*4+3];
    const float bxv= pb[j*4+0], byv= pb[j*4+1], sb = pb[j*4+2], vb = pb[j*4+3];
    const float dxl = bxv - ax, dyl = byv - ay;
    const float nrm = sqrtf(dxl*dxl + dyl*dyl) + 1e-8f;
    const float vx = dxl / nrm, vy = dyl / nrm;

    float cnt = 0.0f, ssum = 0.0f, thr = 0.0f;
    #pragma unroll
    for (int k = 0; k < Ss; ++k) {
      const float t  = (float)k * (1.0f / (float)(Ss - 1));
      const float xs = ax + dxl * t;
      const float ys = ay + dyl * t;
      const int xi = (int)xs, yi = (int)ys;          // trunc-toward-zero == astype(int32)
      const bool inb = (xi >= 0) && (xi < Ww) && (yi >= 0) && (yi < Hh);
      const int xc = min(max(xi, 0), Ww-1);
      const int yc = min(max(yi, 0), Hh-1);
      const int idx = yc * Ww + xc;
      const float vsv = pafx[idx] * vx + pafy[idx] * vy;
      if (inb) { cnt += 1.0f; ssum += vsv; if (vsv > PAF_TH) thr += 1.0f; }
    }
    const float denom = fmaxf(cnt, 1.0f);
    const float mean  = ssum / denom;
    const float ratio = thr  / denom;
    const bool ok = (cnt > 0.0f) && (mean > 0.0f) && (ratio > CNT_TH) &&
                    (va > 0.5f) && (vb > 0.5f);
    out_conn[(size_t)bc * (Pp*Pp) + tid] = ok ? (mean + 0.5f*(sa + sb)) : 0.0f;
  }
}

extern "C" void kernel_launch(void* const* d_in, const int* in_sizes, int n_in,
                              void* d_out, int out_size, void* d_ws, size_t ws_size,
                              hipStream_t stream) {
  (void)in_sizes; (void)n_in; (void)out_size; (void)ws_size;
  const float* heat = (const float*)d_in[0];   // B*K*H*W
  const float* paf  = (const float*)d_in[1];   // B*2C*H*W
  float* out_peaks = (float*)d_out;                 // B*K*P*3 = 32640
  float* out_conn  = out_peaks + Bx*Kk*Pp*3;        // B*C*P*P = 243200
  float* raw = (float*)d_ws;                        // B*K*P*4 scratch (174 KB)

  const size_t shmem = (size_t)NPIX * 2 * sizeof(float);  // 294912 B <= 320 KB/WGP
  (void)hipFuncSetAttribute(reinterpret_cast<const void*>(peaks_kernel),
                            hipFuncAttributeMaxDynamicSharedMemorySize, (int)shmem);

  peaks_kernel<<<Bx*Kk, T1, shmem, stream>>>(heat, out_peaks, raw);
  conn_kernel<<<Bx*Cc, T2, 0, stream>>>(paf, raw, out_conn);
}